// GraphClassifier_54348516163767
// MI455X (gfx1250) — compile-verified
//
#include <hip/hip_runtime.h>

typedef float v2f __attribute__((ext_vector_type(2)));
typedef float v8f __attribute__((ext_vector_type(8)));

#define NN 100000
#define NE 1600000
#define CH 64
#define NG 512

// ---------------------------------------------------------------------------
// H[n,64] = X[n,64] @ W[64,64]   via v_wmma_f32_16x16x4_f32
// Block = 256 threads = 8 waves; each wave owns a 16-row tile (block: 128 rows).
// W staged into LDS (row-major, conflict-free reads: consecutive lanes ->
// consecutive columns).
// ---------------------------------------------------------------------------
__global__ __launch_bounds__(256) void gemm_xw(const float* __restrict__ X,
                                               const float* __restrict__ W,
                                               float* __restrict__ H, int n)
{
    __shared__ float Ws[64 * 64];
    for (int i = threadIdx.x; i < 64 * 64; i += 256) Ws[i] = W[i];
    __syncthreads();

    const int wave = threadIdx.x >> 5;
    const int lane = threadIdx.x & 31;
    const int half = lane >> 4;     // which K-pair of the 4-wide K group
    const int m    = lane & 15;     // row (A) / column (B,C,D) index
    const int rowBase = blockIdx.x * 128 + wave * 16;

    v8f acc0 = {}, acc1 = {}, acc2 = {}, acc3 = {};

    const int rowA  = rowBase + m;
    const int rsafe = rowA < n ? rowA : (n - 1);   // clamp, keep EXEC all-ones

#pragma unroll
    for (int kb = 0; kb < 16; ++kb) {
        const int k0 = kb * 4 + half * 2;
        v2f a = *(const v2f*)(X + (size_t)rsafe * CH + k0);

        v2f b0 = { Ws[(k0 + 0) * 64 +  0 + m], Ws[(k0 + 1) * 64 +  0 + m] };
        v2f b1 = { Ws[(k0 + 0) * 64 + 16 + m], Ws[(k0 + 1) * 64 + 16 + m] };
        v2f b2 = { Ws[(k0 + 0) * 64 + 32 + m], Ws[(k0 + 1) * 64 + 32 + m] };
        v2f b3 = { Ws[(k0 + 0) * 64 + 48 + m], Ws[(k0 + 1) * 64 + 48 + m] };

        acc0 = __builtin_amdgcn_wmma_f32_16x16x4_f32(false, a, false, b0, (short)0, acc0, false, false);
        acc1 = __builtin_amdgcn_wmma_f32_16x16x4_f32(false, a, false, b1, (short)0, acc1, false, false);
        acc2 = __builtin_amdgcn_wmma_f32_16x16x4_f32(false, a, false, b2, (short)0, acc2, false, false);
        acc3 = __builtin_amdgcn_wmma_f32_16x16x4_f32(false, a, false, b3, (short)0, acc3, false, false);
    }

#pragma unroll
    for (int j = 0; j < 8; ++j) {
        int row = rowBase + half * 8 + j;
        if (row < n) {
            float* hr = H + (size_t)row * CH;
            hr[ 0 + m] = acc0[j];
            hr[16 + m] = acc1[j];
            hr[32 + m] = acc2[j];
            hr[48 + m] = acc3[j];
        }
    }
}

// deg[dst[e]] += 1
__global__ __launch_bounds__(256) void deg_kernel(const int* __restrict__ dst,
                                                  float* __restrict__ deg, int e)
{
    int t = blockIdx.x * 256 + threadIdx.x;
    if (t < e) atomicAdd(&deg[dst[t]], 1.0f);
}

// deg -> deg^{-1/2} with self-loop (+1), in place
__global__ __launch_bounds__(256) void dinv_kernel(float* __restrict__ deg, int n)
{
    int t = blockIdx.x * 256 + threadIdx.x;
    if (t < n) deg[t] = rsqrtf(deg[t] + 1.0f);
}

// wave-per-edge: agg[dst] += h[src] * dinv[src]*dinv[dst]  (2 channels/lane)
__global__ __launch_bounds__(256) void edge_scatter(const int* __restrict__ src,
                                                    const int* __restrict__ dst,
                                                    const float* __restrict__ dinv,
                                                    const float* __restrict__ H,
                                                    float* __restrict__ agg, int e)
{
    int gw   = (blockIdx.x * 256 + threadIdx.x) >> 5;
    int lane = threadIdx.x & 31;
    int nw   = (gridDim.x * 256) >> 5;
    for (int ed = gw; ed < e; ed += nw) {
        int s = src[ed], d = dst[ed];
        float nrm = dinv[s] * dinv[d];
        v2f v = *(const v2f*)(H + (size_t)s * CH + lane * 2);
        float* ap = agg + (size_t)d * CH + lane * 2;
        atomicAdd(ap + 0, v.x * nrm);
        atomicAdd(ap + 1, v.y * nrm);
    }
}

// agg += h * dinv^2 + b ; optional ReLU   (one thread per 4 channels)
template <bool RELU>
__global__ __launch_bounds__(256) void selfloop_bias(const float* __restrict__ H,
                                                     const float* __restrict__ dinv,
                                                     const float* __restrict__ b,
                                                     float* __restrict__ agg, int n)
{
    int t = blockIdx.x * 256 + threadIdx.x;
    int node = t >> 4, q = (t & 15) * 4;
    if (node >= n) return;
    float di = dinv[node], w = di * di;
    const float* hp = H + (size_t)node * CH + q;
    float*       ap = agg + (size_t)node * CH + q;
    float4 hv = *(const float4*)hp;
    float4 av = *(const float4*)ap;
    float4 bv = *(const float4*)(b + q);
    float4 r;
    r.x = av.x + hv.x * w + bv.x;
    r.y = av.y + hv.y * w + bv.y;
    r.z = av.z + hv.z * w + bv.z;
    r.w = av.w + hv.w * w + bv.w;
    if (RELU) {
        r.x = fmaxf(r.x, 0.0f); r.y = fmaxf(r.y, 0.0f);
        r.z = fmaxf(r.z, 0.0f); r.w = fmaxf(r.w, 0.0f);
    }
    *(float4*)ap = r;
}

// wave-per-node mean-pool accumulation
__global__ __launch_bounds__(256) void pool_kernel(const float* __restrict__ H,
                                                   const int* __restrict__ batch,
                                                   float* __restrict__ sums,
                                                   float* __restrict__ cnts, int n)
{
    int gw   = (blockIdx.x * 256 + threadIdx.x) >> 5;
    int lane = threadIdx.x & 31;
    int nw   = (gridDim.x * 256) >> 5;
    for (int node = gw; node < n; node += nw) {
        int g = batch[node];
        v2f v = *(const v2f*)(H + (size_t)node * CH + lane * 2);
        float* sp = sums + (size_t)g * CH + lane * 2;
        atomicAdd(sp + 0, v.x);
        atomicAdd(sp + 1, v.y);
        if (lane == 0) atomicAdd(&cnts[g], 1.0f);
    }
}

// wave-per-graph: out[g] = (sums[g] . Wlin) / max(cnt,1) + blin
__global__ __launch_bounds__(256) void head_kernel(const float* __restrict__ sums,
                                                   const float* __restrict__ cnts,
                                                   const float* __restrict__ Wlin,
                                                   const float* __restrict__ blin,
                                                   float* __restrict__ out)
{
    int g    = (blockIdx.x * 256 + threadIdx.x) >> 5;
    int lane = threadIdx.x & 31;
    if (g >= NG) return;
    v2f s = *(const v2f*)(sums + (size_t)g * CH + lane * 2);
    v2f w = *(const v2f*)(Wlin + lane * 2);
    float p = s.x * w.x + s.y * w.y;
#pragma unroll
    for (int off = 16; off > 0; off >>= 1) p += __shfl_xor(p, off, 32);
    if (lane == 0) out[g] = p / fmaxf(cnts[g], 1.0f) + blin[0];
}

extern "C" void kernel_launch(void* const* d_in, const int* in_sizes, int n_in,
                              void* d_out, int out_size, void* d_ws, size_t ws_size,
                              hipStream_t stream)
{
    (void)in_sizes; (void)n_in; (void)out_size; (void)ws_size;
    const float* x    = (const float*)d_in[0];
    const int*   ei   = (const int*)d_in[1];
    const int*   bat  = (const int*)d_in[2];
    const float* W1   = (const float*)d_in[3];
    const float* b1   = (const float*)d_in[4];
    const float* W2   = (const float*)d_in[5];
    const float* b2   = (const float*)d_in[6];
    const float* Wlin = (const float*)d_in[7];
    const float* blin = (const float*)d_in[8];
    float* out = (float*)d_out;

    const int* src = ei;
    const int* dst = ei + NE;

    // workspace partition (256B aligned)
    char* ws = (char*)d_ws;
    size_t off = 0;
    auto carve = [&](size_t bytes) -> void* {
        void* p = ws + off;
        off += (bytes + 255) & ~(size_t)255;
        return p;
    };
    const size_t NHB = (size_t)NN * CH * sizeof(float);   // 25.6 MB
    float* deg  = (float*)carve((size_t)NN * sizeof(float));   // -> dinv
    float* bufA = (float*)carve(NHB);                          // h1, then h2
    float* bufB = (float*)carve(NHB);                          // agg1/hrelu, then agg2
    float* sums = (float*)carve((size_t)NG * CH * sizeof(float));
    float* cnts = (float*)carve((size_t)NG * sizeof(float));

    hipMemsetAsync(deg,  0, (size_t)NN * sizeof(float), stream);
    hipMemsetAsync(bufB, 0, NHB, stream);
    hipMemsetAsync(sums, 0, (size_t)NG * CH * sizeof(float), stream);
    hipMemsetAsync(cnts, 0, (size_t)NG * sizeof(float), stream);

    // degrees -> dinv
    deg_kernel <<<(NE + 255) / 256, 256, 0, stream>>>(dst, deg, NE);
    dinv_kernel<<<(NN + 255) / 256, 256, 0, stream>>>(deg, NN);

    const int gemmBlocks = (NN + 127) / 128;
    const int slBlocks   = (NN * 16 + 255) / 256;

    // ---- layer 1 ----
    gemm_xw<<<gemmBlocks, 256, 0, stream>>>(x, W1, bufA, NN);            // h1 = x@W1
    edge_scatter<<<4096, 256, 0, stream>>>(src, dst, deg, bufA, bufB, NE);
    selfloop_bias<true><<<slBlocks, 256, 0, stream>>>(bufA, deg, b1, bufB, NN); // bufB = relu(agg1)

    // ---- layer 2 ----
    gemm_xw<<<gemmBlocks, 256, 0, stream>>>(bufB, W2, bufA, NN);         // h2 = hrelu@W2
    hipMemsetAsync(bufB, 0, NHB, stream);                                // stream-ordered after GEMM
    edge_scatter<<<4096, 256, 0, stream>>>(src, dst, deg, bufA, bufB, NE);
    selfloop_bias<false><<<slBlocks, 256, 0, stream>>>(bufA, deg, b2, bufB, NN); // bufB = h_out

    // ---- mean pool + linear head ----
    pool_kernel<<<2048, 256, 0, stream>>>(bufB, bat, sums, cnts, NN);
    head_kernel<<<(NG * 32 + 255) / 256, 256, 0, stream>>>(sums, cnts, Wlin, blin, out);
}